// RGATModel_54400055771237
// MI455X (gfx1250) — compile-verified
//
#include <hip/hip_runtime.h>

// ---------------------------------------------------------------------------
// RGAT (3 layers x 3 etypes, H=4, DH=64) for MI455X / gfx1250.
//  * Dense projections + classifier: v_wmma_f32_16x16x4_f32 (fp32 WMMA).
//  * Edge softmax + aggregation: CSR-by-dst (built once per call; the graph is
//    constant across layers), then wave-per-node kernels with ZERO fp atomics.
//    Rationale: the previous atomic formulation needed ~1.84G f32 L2 RMWs
//    (~ms-scale); the CSR form is pure gather + single-writer stores, bounded
//    by ~15 GB of mostly-L2-resident traffic (~0.65 ms at 23.3 TB/s HBM,
//    faster from the 192 MB L2 where z/acc live).
// ---------------------------------------------------------------------------

#define N_NODES   50000
#define E_EDGES   800000
#define ETYPES    3
#define L_LAYERS  3
#define H_HEADS   4
#define DHEAD     64
#define NHID      256          // H_HEADS * DHEAD
#define NCLS      16
#define NEG_SLOPE 0.2f
#define SCAN_THREADS 1024

typedef float v2f __attribute__((ext_vector_type(2)));
typedef float v8f __attribute__((ext_vector_type(8)));

__device__ __forceinline__ float lrelu(float x) {
  return x > 0.f ? x : NEG_SLOPE * x;
}

__global__ void fill_u32(unsigned* __restrict__ p, unsigned v, long n) {
  long i = (long)blockIdx.x * blockDim.x + threadIdx.x;
  if (i < n) p[i] = v;
}

__global__ void copy_i32(const int* __restrict__ a, int* __restrict__ b, int n) {
  int i = blockIdx.x * blockDim.x + threadIdx.x;
  if (i < n) b[i] = a[i];
}

// --------------------------- CSR build (per etype) -------------------------
__global__ void csr_count(const int* __restrict__ dst, int* __restrict__ deg) {
  int e = blockIdx.x * blockDim.x + threadIdx.x;
  if (e < E_EDGES) atomicAdd(&deg[dst[e]], 1);
}

// Single-block exclusive scan over n counts -> rowptr[0..n] (rowptr[n]=total).
__global__ __launch_bounds__(SCAN_THREADS) void exscan_kernel(
    const int* __restrict__ deg, int* __restrict__ rowptr, int n)
{
  __shared__ int ssum[SCAN_THREADS];
  const int t = threadIdx.x;
  const int chunk = (n + SCAN_THREADS - 1) / SCAN_THREADS;
  const int lo = t * chunk;
  const int hi = (lo + chunk < n) ? lo + chunk : n;
  int s = 0;
  for (int i = lo; i < hi; ++i) s += deg[i];
  ssum[t] = s;
  __syncthreads();
  for (int off = 1; off < SCAN_THREADS; off <<= 1) {   // Hillis-Steele
    int tmp = (t >= off) ? ssum[t - off] : 0;
    __syncthreads();
    ssum[t] += tmp;
    __syncthreads();
  }
  int run = ssum[t] - s;                 // exclusive prefix of this chunk
  for (int i = lo; i < hi; ++i) { rowptr[i] = run; run += deg[i]; }
  if (t == SCAN_THREADS - 1) rowptr[n] = run;
}

__global__ void csr_scatter(const int* __restrict__ src,
                            const int* __restrict__ dst,
                            int* __restrict__ cursor, int* __restrict__ col)
{
  int e = blockIdx.x * blockDim.x + threadIdx.x;
  if (e >= E_EDGES) return;
  int pos = atomicAdd(&cursor[dst[e]], 1);
  col[pos] = src[e];                     // only the source id is needed later
}

// ---------------------------------------------------------------------------
// z = h @ W  (50000x256 @ 256x256) with fused attention scores.
// One wave -> 16 rows x 64 cols (one head), v_wmma_f32_16x16x4_f32 over K.
// ---------------------------------------------------------------------------
__global__ __launch_bounds__(256) void gat_gemm_scores(
    const float* __restrict__ hin,   // [N, NHID]
    const float* __restrict__ Wt,    // [NHID, NHID] row-major (K,N)
    const float* __restrict__ alp,   // [H, DHEAD]
    const float* __restrict__ arp,   // [H, DHEAD]
    float* __restrict__ zout,        // [N, NHID]
    float* __restrict__ el,          // [N, H]
    float* __restrict__ er)          // [N, H]
{
  const int lane  = threadIdx.x & 31;
  const int wid   = blockIdx.x * (blockDim.x >> 5) + (threadIdx.x >> 5);
  const int mtile = wid >> 2;
  const int head  = wid & 3;
  if (mtile >= N_NODES / 16) return;   // wave-uniform: EXEC stays all-ones

  const int mbase = mtile * 16;
  const int nbase = head * DHEAD;
  const int l15   = lane & 15;
  const int khalf = (lane >> 4) << 1;  // A/B frag: lanes 0-15 K+{0,1}; 16-31 K+{2,3}
  const int rhalf = (lane >> 4) << 3;  // C layout: +8 rows for upper half-wave

  v8f zero = {0.f, 0.f, 0.f, 0.f, 0.f, 0.f, 0.f, 0.f};
  v8f c[4] = {zero, zero, zero, zero};

  const float* __restrict__ arow = hin + (size_t)(mbase + l15) * NHID;

  for (int k0 = 0; k0 < NHID; k0 += 4) {
    const int kk = k0 + khalf;
    const v2f a = *(const v2f*)(arow + kk);          // global_load_b64
#pragma unroll
    for (int j = 0; j < 4; ++j) {
      const int col = nbase + j * 16 + l15;
      v2f b;
      b.x = Wt[(size_t)kk * NHID + col];
      b.y = Wt[(size_t)(kk + 1) * NHID + col];
      c[j] = __builtin_amdgcn_wmma_f32_16x16x4_f32(
          false, a, false, b, (short)0, c[j], false, false);
    }
  }

#pragma unroll
  for (int i = 0; i < 8; ++i) {
    const int r = mbase + i + rhalf;
#pragma unroll
    for (int j = 0; j < 4; ++j)
      zout[(size_t)r * NHID + nbase + j * 16 + l15] = c[j][i];
  }

  float alv[4], arv[4];
#pragma unroll
  for (int j = 0; j < 4; ++j) {
    alv[j] = alp[head * DHEAD + j * 16 + l15];
    arv[j] = arp[head * DHEAD + j * 16 + l15];
  }
#pragma unroll
  for (int i = 0; i < 8; ++i) {
    float pl = 0.f, pr = 0.f;
#pragma unroll
    for (int j = 0; j < 4; ++j) { pl += c[j][i] * alv[j]; pr += c[j][i] * arv[j]; }
#pragma unroll
    for (int off = 8; off >= 1; off >>= 1) {     // stays inside 16-lane halves
      pl += __shfl_xor(pl, off, 32);
      pr += __shfl_xor(pr, off, 32);
    }
    if (l15 == 0) {
      const int r = mbase + i + rhalf;
      el[r * H_HEADS + head] = pl;
      er[r * H_HEADS + head] = pr;
    }
  }
}

// ---------------------------------------------------------------------------
// Per-dst-node edge softmax, CSR order, no atomics. One wave per node; lanes
// stride the node's in-edges; per-head max/sum via full-wave xor shuffles.
// Writes unnormalized exp() into alphaE (CSR position order) + denom[n,4].
// ---------------------------------------------------------------------------
__global__ __launch_bounds__(256) void gat_softmax_node(
    const int* __restrict__ rowptr, const int* __restrict__ col,
    const float4* __restrict__ el4, const float4* __restrict__ er4,
    float4* __restrict__ alpha4, float4* __restrict__ denom4)
{
  const int lane = threadIdx.x & 31;
  const int node = blockIdx.x * (blockDim.x >> 5) + (threadIdx.x >> 5);
  if (node >= N_NODES) return;
  const int beg = rowptr[node], end = rowptr[node + 1];
  if (beg == end) { if (lane == 0) denom4[node] = make_float4(1.f, 1.f, 1.f, 1.f); return; }
  const float4 rv = er4[node];

  // pass 1: per-head max
  float m0 = -1e30f, m1 = -1e30f, m2 = -1e30f, m3 = -1e30f;
  for (int p = beg + lane; p < end; p += 32) {
    const float4 lv = el4[col[p]];
    m0 = fmaxf(m0, lrelu(lv.x + rv.x));
    m1 = fmaxf(m1, lrelu(lv.y + rv.y));
    m2 = fmaxf(m2, lrelu(lv.z + rv.z));
    m3 = fmaxf(m3, lrelu(lv.w + rv.w));
  }
#pragma unroll
  for (int off = 16; off >= 1; off >>= 1) {
    m0 = fmaxf(m0, __shfl_xor(m0, off, 32));
    m1 = fmaxf(m1, __shfl_xor(m1, off, 32));
    m2 = fmaxf(m2, __shfl_xor(m2, off, 32));
    m3 = fmaxf(m3, __shfl_xor(m3, off, 32));
  }

  // pass 2: exp + per-head sum; stash unnormalized exp in CSR order
  float s0 = 0.f, s1 = 0.f, s2 = 0.f, s3 = 0.f;
  for (int p = beg + lane; p < end; p += 32) {
    const float4 lv = el4[col[p]];
    float4 a;
    a.x = __expf(lrelu(lv.x + rv.x) - m0);
    a.y = __expf(lrelu(lv.y + rv.y) - m1);
    a.z = __expf(lrelu(lv.z + rv.z) - m2);
    a.w = __expf(lrelu(lv.w + rv.w) - m3);
    alpha4[p] = a;
    s0 += a.x; s1 += a.y; s2 += a.z; s3 += a.w;
  }
#pragma unroll
  for (int off = 16; off >= 1; off >>= 1) {
    s0 += __shfl_xor(s0, off, 32);
    s1 += __shfl_xor(s1, off, 32);
    s2 += __shfl_xor(s2, off, 32);
    s3 += __shfl_xor(s3, off, 32);
  }
  if (lane == 0) denom4[node] = make_float4(s0, s1, s2, s3);
}

// ---------------------------------------------------------------------------
// Per-dst-node aggregation, no atomics. One wave per node; lane owns 8
// contiguous columns (a single head). Uniform scalar loads of col/alpha,
// float4 gathers of z[src], one non-atomic RMW of the acc row at the end.
// ---------------------------------------------------------------------------
__global__ __launch_bounds__(256) void gat_node_aggr(
    const int* __restrict__ rowptr, const int* __restrict__ col,
    const float4* __restrict__ alpha4, const float4* __restrict__ denom4,
    const float* __restrict__ z, float* __restrict__ acc)
{
  const int lane = threadIdx.x & 31;
  const int node = blockIdx.x * (blockDim.x >> 5) + (threadIdx.x >> 5);
  if (node >= N_NODES) return;
  const int beg = rowptr[node], end = rowptr[node + 1];
  if (beg == end) return;

  const float4 dv = denom4[node];
  const int c0 = lane * 8;         // this lane's 8 columns, inside one head
  const int hh = lane >> 3;        // head = (lane*8)/64
  const float rcp = 1.f / (hh == 0 ? dv.x : hh == 1 ? dv.y : hh == 2 ? dv.z : dv.w);

  float s0 = 0.f, s1 = 0.f, s2 = 0.f, s3 = 0.f;
  float s4 = 0.f, s5 = 0.f, s6 = 0.f, s7 = 0.f;
  for (int p = beg; p < end; ++p) {            // uniform loop -> scalar loads
    const int sN = col[p];
    const float4 av = alpha4[p];
    const float a = (hh == 0 ? av.x : hh == 1 ? av.y : hh == 2 ? av.z : av.w) * rcp;
    const float4* zr = (const float4*)(z + (size_t)sN * NHID + c0);
    const float4 z0 = zr[0], z1 = zr[1];
    s0 += a * z0.x; s1 += a * z0.y; s2 += a * z0.z; s3 += a * z0.w;
    s4 += a * z1.x; s5 += a * z1.y; s6 += a * z1.z; s7 += a * z1.w;
  }
  float4* ap = (float4*)(acc + (size_t)node * NHID + c0);
  float4 o0 = ap[0], o1 = ap[1];
  o0.x += s0; o0.y += s1; o0.z += s2; o0.w += s3;
  o1.x += s4; o1.y += s5; o1.z += s6; o1.w += s7;
  ap[0] = o0; ap[1] = o1;
}

// ---------------------------------------------------------------------------
// h_next = maybe_relu(acc + sum_t bias[l,t,:])
// ---------------------------------------------------------------------------
__global__ void gat_finish(const float* __restrict__ acc,
                           const float* __restrict__ biasL,  // [ET, NHID]
                           float* __restrict__ hout, int do_relu)
{
  long i = (long)blockIdx.x * blockDim.x + threadIdx.x;
  if (i >= (long)N_NODES * NHID) return;
  const int c = (int)(i & (NHID - 1));
  float v = acc[i] + biasL[c] + biasL[NHID + c] + biasL[2 * NHID + c];
  if (do_relu) v = v > 0.f ? v : 0.f;
  hout[i] = v;
}

// ---------------------------------------------------------------------------
// out = h @ lin_w + lin_b   ([N,256]x[256,16]) — single WMMA N-tile per wave.
// ---------------------------------------------------------------------------
__global__ __launch_bounds__(256) void gat_out_linear(
    const float* __restrict__ hin, const float* __restrict__ lw,
    const float* __restrict__ lb, float* __restrict__ out)
{
  const int lane = threadIdx.x & 31;
  const int wid  = blockIdx.x * (blockDim.x >> 5) + (threadIdx.x >> 5);
  if (wid >= N_NODES / 16) return;
  const int mbase = wid * 16;
  const int l15   = lane & 15;
  const int khalf = (lane >> 4) << 1;
  const int rhalf = (lane >> 4) << 3;

  v8f c = {0.f, 0.f, 0.f, 0.f, 0.f, 0.f, 0.f, 0.f};
  const float* __restrict__ arow = hin + (size_t)(mbase + l15) * NHID;

  for (int k0 = 0; k0 < NHID; k0 += 4) {
    const int kk = k0 + khalf;
    const v2f a = *(const v2f*)(arow + kk);
    v2f b;
    b.x = lw[kk * NCLS + l15];
    b.y = lw[(kk + 1) * NCLS + l15];
    c = __builtin_amdgcn_wmma_f32_16x16x4_f32(
        false, a, false, b, (short)0, c, false, false);
  }

  const float bb = lb[l15];
#pragma unroll
  for (int i = 0; i < 8; ++i) {
    const int r = mbase + i + rhalf;
    out[r * NCLS + l15] = c[i] + bb;
  }
}

// ---------------------------------------------------------------------------
extern "C" void kernel_launch(void* const* d_in, const int* in_sizes, int n_in,
                              void* d_out, int out_size, void* d_ws, size_t ws_size,
                              hipStream_t stream)
{
  const float* x      = (const float*)d_in[0];
  const int*   src    = (const int*)d_in[1];   // [ET, E]
  const int*   dst    = (const int*)d_in[2];   // [ET, E]
  const float* W      = (const float*)d_in[3]; // [L, ET, NHID, NHID]
  const float* attn_l = (const float*)d_in[4]; // [L, ET, H, DHEAD]
  const float* attn_r = (const float*)d_in[5];
  const float* bias   = (const float*)d_in[6]; // [L, ET, NHID]
  const float* lin_w  = (const float*)d_in[7]; // [NHID, NCLS]
  const float* lin_b  = (const float*)d_in[8]; // [NCLS]
  float* out = (float*)d_out;
  (void)in_sizes; (void)n_in; (void)out_size; (void)ws_size;

  const size_t NNH = (size_t)N_NODES * NHID;     // 12.8M floats
  const size_t NHh = (size_t)N_NODES * H_HEADS;  // 200K floats
  const size_t EH  = (size_t)E_EDGES * H_HEADS;  // 3.2M floats

  float* wsf   = (float*)d_ws;
  float* hbuf  = wsf;                     // [N, NHID]
  float* acc   = hbuf + NNH;              // [N, NHID]
  float* zbuf  = acc + NNH;               // [N, NHID]
  float* el    = zbuf + NNH;              // [N, H]
  float* er    = el + NHh;                // [N, H]
  float* denom = er + NHh;                // [N, H]
  float* alphaE= denom + NHh;             // [E, H] (CSR position order)
  int*   rowptr= (int*)(alphaE + EH);     // [ET][N+1]
  int*   colb  = rowptr + (size_t)ETYPES * (N_NODES + 1);  // [ET][E]
  int*   deg   = colb + (size_t)ETYPES * E_EDGES;          // [N] temp
  int*   cursor= deg + N_NODES;                            // [N] temp

  const dim3 blk(256);
  const int MT          = N_NODES / 16;              // 3125 row tiles (exact)
  const int gemm_blocks = (MT * H_HEADS + 7) / 8;
  const int edge_blocks = (E_EDGES + 255) / 256;
  const int node_blocks = (N_NODES + 7) / 8;         // wave per node
  const int nnh_blocks  = (int)((NNH + 255) / 256);
  const int n_blocks    = (N_NODES + 255) / 256;

  // ---- CSR build: once per call; the graph is shared by all 3 layers ----
  for (int t = 0; t < ETYPES; ++t) {
    const int* st = src + (size_t)t * E_EDGES;
    const int* dt = dst + (size_t)t * E_EDGES;
    int* rp = rowptr + (size_t)t * (N_NODES + 1);
    int* cl = colb + (size_t)t * E_EDGES;
    fill_u32<<<n_blocks, blk, 0, stream>>>((unsigned*)deg, 0u, (long)N_NODES);
    csr_count<<<edge_blocks, blk, 0, stream>>>(dt, deg);
    exscan_kernel<<<1, SCAN_THREADS, 0, stream>>>(deg, rp, N_NODES);
    copy_i32<<<n_blocks, blk, 0, stream>>>(rp, cursor, N_NODES);
    csr_scatter<<<edge_blocks, blk, 0, stream>>>(st, dt, cursor, cl);
  }

  // ---- layers ----
  const float* hcur = x;
  for (int l = 0; l < L_LAYERS; ++l) {
    fill_u32<<<nnh_blocks, blk, 0, stream>>>((unsigned*)acc, 0u, (long)NNH);
    for (int t = 0; t < ETYPES; ++t) {
      const float* Wt  = W + ((size_t)l * ETYPES + t) * NHID * NHID;
      const float* alp = attn_l + ((size_t)l * ETYPES + t) * NHID;
      const float* arp = attn_r + ((size_t)l * ETYPES + t) * NHID;
      const int* rp = rowptr + (size_t)t * (N_NODES + 1);
      const int* cl = colb + (size_t)t * E_EDGES;

      gat_gemm_scores<<<gemm_blocks, blk, 0, stream>>>(hcur, Wt, alp, arp,
                                                       zbuf, el, er);
      gat_softmax_node<<<node_blocks, blk, 0, stream>>>(
          rp, cl, (const float4*)el, (const float4*)er,
          (float4*)alphaE, (float4*)denom);
      gat_node_aggr<<<node_blocks, blk, 0, stream>>>(
          rp, cl, (const float4*)alphaE, (const float4*)denom, zbuf, acc);
    }
    gat_finish<<<nnh_blocks, blk, 0, stream>>>(
        acc, bias + (size_t)l * ETYPES * NHID, hbuf, l != L_LAYERS - 1);
    hcur = hbuf;
  }
  gat_out_linear<<<(MT + 7) / 8, blk, 0, stream>>>(hcur, lin_w, lin_b, out);
}